// updata_z_18811956756794
// MI455X (gfx1250) — compile-verified
//
#include <hip/hip_runtime.h>

// ---------------------------------------------------------------------------
// MRI data-consistency update via DFT-as-GEMM on CDNA5 WMMA (gfx1250, wave32).
//
//   z = | IFFT2( (1-A) .* FFT2(x) + y_k ) |
//
// FFT2(x)  = F @ x @ F          (F = 240x240 DFT matrix, symmetric)
// IFFT2(z) = (1/N^2) conj(F) @ z @ conj(F)
//
// 240 = 15*16 -> exact 16x16 WMMA tiling; K stepped by 4 with
// v_wmma_f32_16x16x4_f32 (fp32 matrix op keeps DFT numerics exact-ish).
//
// Blocked complex GEMM:
//   block = 256 threads = 8 waves, tile = 32(M) x 64(N) (2x4 wave tiles)
//   K chunked at 48 (240 = 5*48); A panel (32x48x2) and B panel (48x64x2)
//   staged in LDS with coalesced b128 global loads; fragments fed from LDS
//   (ds_load) with bank-conflict-free padded strides (A:52, B:72 dwords).
//   Edge tiles handled by wave-uniform predicates (EXEC all-1s for WMMA);
//   all waves keep participating in staging + barriers.
// ---------------------------------------------------------------------------

#define HW     240
#define NPIX   (HW * HW)            // 57600
#define NBATCH 256
#define PLANE  ((size_t)NBATCH * NPIX)

#define KC     48                   // K chunk (240 = 5*48)
#define ASTR   52                   // A LDS row stride (dwords), bank-safe, 16B rows
#define BSTR   72                   // B LDS row stride (dwords), bank-safe, 16B rows
#define A_PLANE (32 * ASTR)         // 1664 dwords per plane
#define B_PLANE (KC * BSTR)         // 3456 dwords per plane

typedef float v2f __attribute__((ext_vector_type(2)));
typedef float v4f __attribute__((ext_vector_type(4)));
typedef float v8f __attribute__((ext_vector_type(8)));

#define WMMA_F32(A, B, C) \
  __builtin_amdgcn_wmma_f32_16x16x4_f32(false, (A), false, (B), (short)0, (C), false, false)

// --- DFT twiddle build -----------------------------------------------------
__global__ void build_dft_kernel(float* __restrict__ Fr, float* __restrict__ Fi) {
  int idx = blockIdx.x * blockDim.x + threadIdx.x;
  if (idx >= NPIX) return;
  int j = idx / HW, k = idx % HW;
  int m = (j * k) % HW;                       // phase reduced mod N for accuracy
  double ang = -2.0 * 3.14159265358979323846 * (double)m / (double)HW;
  Fr[idx] = (float)cos(ang);
  Fi[idx] = (float)sin(ang);
}

// --- blocked complex GEMM with fused epilogues -----------------------------
// C = (P + sq*Q) + i*(sr*R + ss*S), P=Ar*Br, Q=Ai*Bi, R=Ar*Bi, S=Ai*Br
// MODE 0: store Cr,Ci
// MODE 1: data consistency: C' = (1-mask)*C + y_k  (y_k = Y[b][0] + i*Y[b][1])
// MODE 2: magnitude: out = scale * |C|
// BREAL : B has no imaginary plane (stage 1), Q=R=0.
template <int MODE, bool BREAL>
__global__ void __launch_bounds__(256)
cgemm_blocked(const float* __restrict__ Ar, const float* __restrict__ Ai, int strideA,
              const float* __restrict__ Br, const float* __restrict__ Bi, int strideB,
              float sq, float sr, float ss,
              const float* __restrict__ mask, const float* __restrict__ Y,
              float scale,
              float* __restrict__ Cr, float* __restrict__ Ci, int strideC) {
  __shared__ float sA[2 * A_PLANE];           // 13312 B
  __shared__ float sB[2 * B_PLANE];           // 27648 B

  const int tid  = threadIdx.x;
  const int lane = tid & 31;
  const int wave = tid >> 5;                  // 0..7
  const int mSub = wave >> 2;                 // 0..1
  const int nSub = wave & 3;                  // 0..3
  const int half = lane >> 4;
  const int l    = lane & 15;

  const int mBase = blockIdx.x * 32;          // 8 groups -> rows [0,256)
  const int nBase = blockIdx.y * 64;          // 4 groups -> cols [0,256)
  const int mTile = blockIdx.x * 2 + mSub;    // 16x16 tile indices
  const int nTile = blockIdx.y * 4 + nSub;
  const bool valid = (mTile < HW / 16) && (nTile < HW / 16);  // wave-uniform

  const int b = blockIdx.z;
  const float* ar_p = Ar + (size_t)b * strideA;
  const float* ai_p = Ai + (size_t)b * strideA;
  const float* br_p = Br + (size_t)b * strideB;
  const float* bi_p = BREAL ? nullptr : (Bi + (size_t)b * strideB);

  v8f P = {}; v8f Q = {}; v8f R = {}; v8f S = {};

  for (int kc = 0; kc < HW; kc += KC) {
    // ---- stage A panel: [mBase, mBase+32) x [kc, kc+KC), 2 planes ----
    // 2*32*12 = 768 float4, 256 threads -> 3 each (coalesced 128b rows)
#pragma unroll
    for (int i = 0; i < 3; ++i) {
      int idx   = tid + i * 256;
      int plane = idx / 384;
      int rem   = idx % 384;
      int row   = rem / 12;
      int c4    = (rem % 12) * 4;
      const float* src = plane ? ai_p : ar_p;
      int gRow = mBase + row;
      v4f v = {};
      if (gRow < HW)
        v = *reinterpret_cast<const v4f*>(src + (size_t)gRow * HW + kc + c4);
      *reinterpret_cast<v4f*>(&sA[plane * A_PLANE + row * ASTR + c4]) = v;
    }
    // ---- stage B panel: [kc, kc+KC) x [nBase, nBase+64), 1-2 planes ----
    // per plane 48*16 = 768 float4 -> 3 iterations each
    constexpr int NITB = BREAL ? 3 : 6;
#pragma unroll
    for (int i = 0; i < NITB; ++i) {
      int idx   = tid + i * 256;
      int plane = idx / 768;
      int rem   = idx % 768;
      int row   = rem / 16;
      int c4    = (rem % 16) * 4;
      const float* src = plane ? bi_p : br_p;
      int gCol = nBase + c4;
      v4f v = {};
      if (gCol < HW)
        v = *reinterpret_cast<const v4f*>(src + (size_t)(kc + row) * HW + gCol);
      *reinterpret_cast<v4f*>(&sB[plane * B_PLANE + row * BSTR + c4]) = v;
    }
    __syncthreads();

    // ---- WMMA inner loop over the chunk (EXEC all-1s inside 'valid') ----
    if (valid) {
#pragma unroll
      for (int kk = 0; kk < KC; kk += 4) {
        // A frag (16x4): lanes 0-15 row M=l K={kk,kk+1}; lanes 16-31 K={kk+2,kk+3}
        const int aoff = (mSub * 16 + l) * ASTR + kk + 2 * half;
        v2f ar = *reinterpret_cast<const v2f*>(&sA[aoff]);
        v2f ai = *reinterpret_cast<const v2f*>(&sA[A_PLANE + aoff]);
        // B frag (4x16): v0 rows {kk, kk+2}; v1 rows {kk+1, kk+3}
        const int boff = (kk + 2 * half) * BSTR + nSub * 16 + l;
        v2f br; br.x = sB[boff]; br.y = sB[boff + BSTR];
        P = WMMA_F32(ar, br, P);
        S = WMMA_F32(ai, br, S);
        if (!BREAL) {
          v2f bi; bi.x = sB[B_PLANE + boff]; bi.y = sB[B_PLANE + boff + BSTR];
          Q = WMMA_F32(ai, bi, Q);
          R = WMMA_F32(ar, bi, R);
        }
      }
    }
    __syncthreads();
  }

  if (!valid) return;

  // ---- epilogue: C layout = VGPR r -> row mTile*16 + r + 8*half, col nTile*16 + l
#pragma unroll
  for (int r = 0; r < 8; ++r) {
    size_t off = (size_t)(mTile * 16 + r + 8 * half) * HW + nTile * 16 + l;
    float cr = BREAL ? P[r] : (P[r] + sq * Q[r]);
    float ci = BREAL ? S[r] : (sr * R[r] + ss * S[r]);
    if (MODE == 1) {
      float a = 1.0f - mask[off];
      cr = a * cr + Y[(size_t)b * 2 * NPIX + off];
      ci = a * ci + Y[(size_t)b * 2 * NPIX + NPIX + off];
      Cr[(size_t)b * strideC + off] = cr;
      Ci[(size_t)b * strideC + off] = ci;
    } else if (MODE == 2) {
      Cr[(size_t)b * strideC + off] = scale * sqrtf(cr * cr + ci * ci);
    } else {
      Cr[(size_t)b * strideC + off] = cr;
      Ci[(size_t)b * strideC + off] = ci;
    }
  }
}

// ---------------------------------------------------------------------------
extern "C" void kernel_launch(void* const* d_in, const int* in_sizes, int n_in,
                              void* d_out, int out_size, void* d_ws, size_t ws_size,
                              hipStream_t stream) {
  const float* X   = (const float*)d_in[0];   // T2_Gen   [256,1,240,240]
  const float* Y   = (const float*)d_in[1];   // underT2_K[256,2,240,240]
  const float* Msk = (const float*)d_in[2];   // A        [240,240]
  float*       out = (float*)d_out;           //          [256,1,240,240]

  // Workspace: Fr, Fi, then two ping-pong complex batch buffers.
  float* w   = (float*)d_ws;
  float* Fr  = w;
  float* Fi  = Fr + NPIX;
  float* B0r = Fi + NPIX;
  float* B0i = B0r + PLANE;
  float* B1r = B0i + PLANE;
  float* B1i = B1r + PLANE;

  build_dft_kernel<<<(NPIX + 255) / 256, 256, 0, stream>>>(Fr, Fi);

  dim3 grid(8, 4, NBATCH);   // 32x64 block tiles over 240x240, z = batch
  dim3 block(256);           // 8 waves (wave32)

  // 1) T = F @ x                    (B real: Cr=P, Ci=S)
  cgemm_blocked<0, true><<<grid, block, 0, stream>>>(
      Fr, Fi, 0, X, nullptr, NPIX,
      0.0f, 0.0f, 1.0f, nullptr, nullptr, 0.0f,
      B0r, B0i, NPIX);

  // 2) Zk = (1-A).*(T @ F) + y_k    (normal x normal: Cr=P-Q, Ci=R+S)
  cgemm_blocked<1, false><<<grid, block, 0, stream>>>(
      B0r, B0i, NPIX, Fr, Fi, 0,
      -1.0f, 1.0f, 1.0f, Msk, Y, 0.0f,
      B1r, B1i, NPIX);

  // 3) U = conj(F) @ Zk             (conj-A: Cr=P+Q, Ci=R-S)
  cgemm_blocked<0, false><<<grid, block, 0, stream>>>(
      Fr, Fi, 0, B1r, B1i, NPIX,
      1.0f, 1.0f, -1.0f, nullptr, nullptr, 0.0f,
      B0r, B0i, NPIX);

  // 4) out = |U @ conj(F)| / N^2    (conj-B: Cr=P+Q, Ci=-R+S)
  cgemm_blocked<2, false><<<grid, block, 0, stream>>>(
      B0r, B0i, NPIX, Fr, Fi, 0,
      1.0f, -1.0f, 1.0f, nullptr, nullptr, 1.0f / (float)NPIX,
      out, nullptr, NPIX);
}